// GCNLayer_21492016349947
// MI455X (gfx1250) — compile-verified
//
#include <hip/hip_runtime.h>
#include <math.h>

typedef float v2f __attribute__((ext_vector_type(2)));
typedef float v8f __attribute__((ext_vector_type(8)));

#define DFEAT 128
#define LN_EPS 1e-5f

// ---------------- Kernel 1: in-degree via float atomics ----------------
__global__ __launch_bounds__(256) void k_degree(const long long* __restrict__ ei,
                                                float* __restrict__ deg, int E) {
  int t = blockIdx.x * blockDim.x + threadIdx.x;
  if (t < E) {
    int dst = (int)ei[(size_t)E + t];
    atomicAdd(&deg[dst], 1.0f);
  }
}

// ---------------- Kernel 2: deg_inv = (deg+1)^-0.5 (in place) ----------
__global__ __launch_bounds__(256) void k_deginv(float* __restrict__ deg, int N) {
  int t = blockIdx.x * blockDim.x + threadIdx.x;
  if (t < N) deg[t] = rsqrtf(deg[t] + 1.0f);
}

// ---------------- Kernel 3: pack W into k-pair interleaved layout ------
// Wp[kp*128 + n] = (W[2kp][n], W[2kp+1][n])  -> matches WMMA B operand
__global__ __launch_bounds__(256) void k_packW(const float* __restrict__ W,
                                               float2* __restrict__ Wp) {
  int t = blockIdx.x * blockDim.x + threadIdx.x;
  if (t < 64 * 128) {
    int kp = t >> 7, n = t & 127;
    Wp[t] = make_float2(W[(size_t)(2 * kp) * DFEAT + n],
                        W[(size_t)(2 * kp + 1) * DFEAT + n]);
  }
}

// ---------------- Kernel 4: edge scatter (one wave per edge) -----------
__global__ __launch_bounds__(256) void k_scatter(const long long* __restrict__ ei,
                                                 const float* __restrict__ ew,
                                                 const float* __restrict__ x,
                                                 const float* __restrict__ dinv,
                                                 float* __restrict__ agg, int E) {
  int t = blockIdx.x * blockDim.x + threadIdx.x;
  int e = t >> 5;
  int lane = t & 31;
  if (e >= E) return;
  long long src = ei[e];
  long long dst = ei[(size_t)E + e];
  float s = dinv[src] * ew[e];
  const float4* xr = (const float4*)(x + (size_t)src * DFEAT);
  float4 v = xr[lane];                    // features lane*4 .. lane*4+3
  float* ar = agg + (size_t)dst * DFEAT + lane * 4;
  atomicAdd(ar + 0, v.x * s);
  atomicAdd(ar + 1, v.y * s);
  atomicAdd(ar + 2, v.z * s);
  atomicAdd(ar + 3, v.w * s);
}

// ---------------- Kernel 5: fused finalize + GEMM(WMMA) + GELU + LN ----
__global__ __launch_bounds__(256) void k_fused(const float* __restrict__ agg,
                                               const float* __restrict__ x,
                                               const float* __restrict__ dinv,
                                               const float2* __restrict__ Wp,
                                               const float* __restrict__ bias,
                                               const float* __restrict__ gamma,
                                               const float* __restrict__ beta,
                                               float* __restrict__ out, int N) {
  const int wave = threadIdx.x >> 5;
  const int lane = threadIdx.x & 31;
  const int half = lane >> 4;     // 0: K lo pair / rows 0-7 ; 1: K hi pair / rows 8-15
  const int m16  = lane & 15;
  const int base = (blockIdx.x * 8 + wave) * 16;   // 16-node tile per wave
  if (base >= N) return;                            // wave-uniform exit

  int arow_i = base + m16;
  if (arow_i >= N) arow_i = N - 1;                  // clamp; stores are guarded
  const float di = dinv[arow_i];
  const float2* arow = (const float2*)(agg + (size_t)arow_i * DFEAT);
  const float2* xrow = (const float2*)(x   + (size_t)arow_i * DFEAT);

  const v8f vzero = {0.f, 0.f, 0.f, 0.f, 0.f, 0.f, 0.f, 0.f};
  v8f acc[8];
#pragma unroll
  for (int i = 0; i < 8; ++i) acc[i] = vzero;

  // D[16x128] = A[16x128] * W[128x128], K swept 4 at a time
#pragma unroll 4
  for (int t = 0; t < 32; ++t) {
    const int kp = 2 * t + half;                    // k-pair index for this half
    float2 av = arow[kp];
    float2 xv = xrow[kp];
    v2f a; a.x = av.x * di + xv.x;                  // agg*deg_inv + x (residual)
           a.y = av.y * di + xv.y;
    const float2* wr = Wp + (size_t)kp * DFEAT + m16;
#pragma unroll
    for (int nt = 0; nt < 8; ++nt) {
      float2 bv = wr[nt * 16];
      v2f b; b.x = bv.x; b.y = bv.y;
      acc[nt] = __builtin_amdgcn_wmma_f32_16x16x4_f32(
          false, a, false, b, (short)0, acc[nt], false, false);
    }
  }

  // bias + exact GELU (erf form), done in-place on accumulators
  float gam[8], bet[8];
#pragma unroll
  for (int nt = 0; nt < 8; ++nt) {
    const int n = nt * 16 + m16;
    const float bn = bias[n];
    gam[nt] = gamma[n];
    bet[nt] = beta[n];
#pragma unroll
    for (int r = 0; r < 8; ++r) {
      float v = acc[nt][r] + bn;
      acc[nt][r] = 0.5f * v * (1.0f + erff(v * 0.70710678118654752f));
    }
  }

  // LayerNorm stats: VGPR r holds row (r + 8*half); reduce across 16 lanes of the half
  float s[8], ss[8];
#pragma unroll
  for (int r = 0; r < 8; ++r) {
    float a0 = 0.f, a1 = 0.f;
#pragma unroll
    for (int nt = 0; nt < 8; ++nt) { float v = acc[nt][r]; a0 += v; a1 += v * v; }
    s[r] = a0; ss[r] = a1;
  }
#pragma unroll
  for (int mask = 1; mask < 16; mask <<= 1) {
#pragma unroll
    for (int r = 0; r < 8; ++r) {
      s[r]  += __shfl_xor(s[r],  mask, 32);
      ss[r] += __shfl_xor(ss[r], mask, 32);
    }
  }

#pragma unroll
  for (int r = 0; r < 8; ++r) {
    const int node = base + r + 8 * half;
    if (node >= N) continue;
    const float mu  = s[r]  * (1.0f / 128.0f);
    const float var = ss[r] * (1.0f / 128.0f) - mu * mu;
    const float rs  = rsqrtf(var + LN_EPS);
    float* orow = out + (size_t)node * DFEAT;
#pragma unroll
    for (int nt = 0; nt < 8; ++nt)
      orow[nt * 16 + m16] = (acc[nt][r] - mu) * rs * gam[nt] + bet[nt];
  }
}

// ---------------------------------------------------------------------
extern "C" void kernel_launch(void* const* d_in, const int* in_sizes, int n_in,
                              void* d_out, int out_size, void* d_ws, size_t ws_size,
                              hipStream_t stream) {
  const float*     x     = (const float*)d_in[0];
  const long long* ei    = (const long long*)d_in[1];
  const float*     ew    = (const float*)d_in[2];
  const float*     W     = (const float*)d_in[3];
  const float*     bias  = (const float*)d_in[4];
  const float*     gamma = (const float*)d_in[5];
  const float*     beta  = (const float*)d_in[6];
  float*           out   = (float*)d_out;

  const int N = in_sizes[0] / DFEAT;
  const int E = in_sizes[2];

  char* ws = (char*)d_ws;
  size_t degBytes = ((size_t)N * sizeof(float) + 255) & ~(size_t)255;
  size_t aggBytes = (size_t)N * DFEAT * sizeof(float);
  size_t aggPad   = (aggBytes + 255) & ~(size_t)255;
  float*  deg = (float*)ws;                  // N floats (becomes deg_inv)
  float*  agg = (float*)(ws + degBytes);     // N*128 floats
  float2* Wp  = (float2*)(ws + degBytes + aggPad);  // 64*128 float2

  // zero deg + agg each call (deterministic; Wp fully overwritten)
  hipMemsetAsync(ws, 0, degBytes + aggBytes, stream);

  k_degree<<<(E + 255) / 256, 256, 0, stream>>>(ei, deg, E);
  k_deginv<<<(N + 255) / 256, 256, 0, stream>>>(deg, N);
  k_packW <<<(64 * 128 + 255) / 256, 256, 0, stream>>>(W, Wp);

  long long scatterThreads = (long long)E * 32;
  k_scatter<<<(int)((scatterThreads + 255) / 256), 256, 0, stream>>>(ei, ew, x, deg, agg, E);

  k_fused<<<(N + 127) / 128, 256, 0, stream>>>(agg, x, deg, Wp, bias, gamma, beta, out, N);
}